// DimeNetPP_79542794322610
// MI455X (gfx1250) — compile-verified
//
#include <hip/hip_runtime.h>
#include <hip/hip_bf16.h>

#define N_EDGES 2000000
#define N_TRIP  6000000

typedef __attribute__((ext_vector_type(2))) float v2f;
typedef __attribute__((ext_vector_type(8))) float v8f;

// Spherical Bessel zeros z_{l,n}, l=0..6, n=0..5 (standard tables)
__constant__ float ZT[7][6] = {
  { 3.1415927f,  6.2831853f,  9.4247780f, 12.5663706f, 15.7079633f, 18.8495559f },
  { 4.4934095f,  7.7252518f, 10.9041217f, 14.0661939f, 17.2207553f, 20.3713030f },
  { 5.7634592f,  9.0950113f, 12.3229410f, 15.5146030f, 18.6890364f, 21.8538742f },
  { 6.9879320f, 10.4171186f, 13.6980232f, 16.9236212f, 20.1218061f, 23.3042470f },
  { 8.1825615f, 11.7049071f, 15.0396647f, 18.3012559f, 21.5254177f, 24.7275655f },
  { 9.3558121f, 12.9665301f, 16.3547097f, 19.6531521f, 22.9045507f, 26.1277501f },
  {10.5128354f, 14.2073924f, 17.6479748f, 20.9834630f, 24.2627680f, 27.5078684f }
};

// sqrt((2l+1)/(4pi))
#define SN0 0.2820947918f
#define SN1 0.4886025119f
#define SN2 0.6307831305f
#define SN3 0.7463526652f
#define SN4 0.8462843753f
#define SN5 0.9356025796f
#define SN6 1.0171072363f

// B-matrix for WMMA: BM[k][l] = coeff of c^k in SN[l]*P_l(c); cols 7..15 = 0, row 7 = 0
__constant__ float BM[8][16] = {
  { SN0, 0.0f, -0.5f*SN2, 0.0f,  0.375f*SN4, 0.0f,  -0.3125f*SN6, 0,0,0,0,0,0,0,0,0 },
  { 0.0f, SN1, 0.0f, -1.5f*SN3,  0.0f, 1.875f*SN5,   0.0f,        0,0,0,0,0,0,0,0,0 },
  { 0.0f, 0.0f, 1.5f*SN2, 0.0f, -3.75f*SN4, 0.0f,    6.5625f*SN6, 0,0,0,0,0,0,0,0,0 },
  { 0.0f, 0.0f, 0.0f, 2.5f*SN3,  0.0f, -8.75f*SN5,   0.0f,        0,0,0,0,0,0,0,0,0 },
  { 0.0f, 0.0f, 0.0f, 0.0f, 4.375f*SN4, 0.0f, -19.6875f*SN6,      0,0,0,0,0,0,0,0,0 },
  { 0.0f, 0.0f, 0.0f, 0.0f, 0.0f, 7.875f*SN5,  0.0f,              0,0,0,0,0,0,0,0,0 },
  { 0.0f, 0.0f, 0.0f, 0.0f, 0.0f, 0.0f, 14.4375f*SN6,             0,0,0,0,0,0,0,0,0 },
  { 0,0,0,0,0,0,0,0,0,0,0,0,0,0,0,0 }
};

// Compute norm[l*6+n] = sqrt(2)/|j_{l+1}(z_{l,n})| in double precision (42 values).
__global__ void norm_kernel(float* __restrict__ normw) {
  int idx = threadIdx.x;
  if (idx >= 42) return;
  int l = idx / 6, n = idx - 6 * l;
  double z = (double)ZT[l][n];
  double s = sin(z), co = cos(z);
  double j0 = s / z;
  double j1 = s / (z * z) - co / z;
  double jm = j0, jc = j1;
  for (int t = 1; t <= l; ++t) { double nx = (2.0 * t + 1.0) / z * jc - jm; jm = jc; jc = nx; }
  normw[idx] = (float)(sqrt(2.0) / fabs(jc));   // jc == j_{l+1}(z)
}

// Per-edge: dist_embs -> d_out[0 : E*6], rbf (env*norm*j_l) -> workspace [E][42],
// staged through LDS so the 336MB workspace write is fully coalesced.
__global__ __launch_bounds__(256)
void edge_kernel(const float* __restrict__ coords, const float* __restrict__ freq,
                 const int* __restrict__ nis, const int* __restrict__ njs,
                 const float* __restrict__ normw,
                 float* __restrict__ dist_out, float* __restrict__ rbf_ws) {
  __shared__ float lds[256 * 42];
  int tid  = threadIdx.x;
  int base = blockIdx.x * 256;
  int e    = base + tid;
  bool act = e < N_EDGES;
  int ee   = act ? e : (N_EDGES - 1);

  int i = nis[ee], j = njs[ee];
  float dx = coords[3*i]   - coords[3*j];
  float dy = coords[3*i+1] - coords[3*j+1];
  float dz = coords[3*i+2] - coords[3*j+2];
  float d  = sqrtf(dx*dx + dy*dy + dz*dz);
  float x  = d * 0.2f;                              // d / CUTOFF
  float x2 = x * x;
  float x5 = x2 * x2 * x;
  float env = 1.0f / x + x5 * (-28.0f + x * (48.0f + x * 21.0f));  // reference envelope(p=5)

  if (act) {
    #pragma unroll
    for (int n = 0; n < 6; ++n)
      dist_out[(size_t)e * 6 + n] = env * sinf(freq[n] * x);
  }

  #pragma unroll
  for (int l = 0; l < 7; ++l) {
    #pragma unroll
    for (int n = 0; n < 6; ++n) {
      float arg = ZT[l][n] * x;
      float s, c;
      sincosf(arg, &s, &c);
      float ia = 1.0f / arg;
      float j0 = s * ia;
      float jv;
      if (l == 0) {
        jv = j0;
      } else {
        float j1 = s * ia * ia - c * ia;
        float jm = j0, jc = j1;
        #pragma unroll
        for (int q = 1; q < l; ++q) { float nx = (2.0f*q + 1.0f) * ia * jc - jm; jm = jc; jc = nx; }
        jv = jc;
      }
      lds[tid * 42 + l * 6 + n] = env * normw[l * 6 + n] * jv;
    }
  }
  __syncthreads();

  int count = min(256, N_EDGES - base);
  int total = count * 42;
  size_t gbase = (size_t)base * 42;
  #pragma unroll
  for (int it = 0; it < 42; ++it) {
    int flat = it * 256 + tid;
    if (flat < total) rbf_ws[gbase + flat] = lds[flat];   // fully coalesced
  }
}

// Per-triplet: cos(theta) analytically, cbf via V_WMMA_F32_16X16X4_F32 (powers-of-c x
// Legendre coeff matrix), then gather rbf[kj] and emit angle_embs with coalesced stores.
// One wave = one 16-triplet WMMA tile; lanes L and L+16 handle the same triplet (A K-split).
__global__ __launch_bounds__(256)
void trip_kernel(const float* __restrict__ coords,
                 const int* __restrict__ trip_i, const int* __restrict__ trip_j,
                 const int* __restrict__ trip_k, const int* __restrict__ kj,
                 const float* __restrict__ rbf_ws, float* __restrict__ angle_out) {
  __shared__ float cbf_s[8][16][8];
  __shared__ int   kj_s[8][16];

  int tid  = threadIdx.x;
  int lane = tid & 31;
  int wave = tid >> 5;
  int hi   = lane >> 4;      // 0: K-low half, 1: K-high half
  int tl   = lane & 15;      // triplet index within tile (row M)
  int t0   = blockIdx.x * 128 + wave * 16;
  int t    = t0 + tl;        // N_TRIP % 128 == 0 -> never OOB, EXEC stays all-ones

  int ti = trip_i[t], tj = trip_j[t], tk = trip_k[t];
  int e  = kj[t];
  __builtin_prefetch(rbf_ws + (size_t)e * 42, 0, 1);   // global_prefetch_b8: hide gather latency

  float pjx = coords[3*tj], pjy = coords[3*tj+1], pjz = coords[3*tj+2];
  float ax = coords[3*ti]   - pjx, ay = coords[3*ti+1] - pjy, az = coords[3*ti+2] - pjz;
  float bx = coords[3*tk]   - pjx, by = coords[3*tk+1] - pjy, bz = coords[3*tk+2] - pjz;
  float dt = ax*bx + ay*by + az*bz;
  float cx = ay*bz - az*by, cy = az*bx - ax*bz, cz = ax*by - ay*bx;
  float b2 = cx*cx + cy*cy + cz*cz;
  float c  = dt * rsqrtf(dt*dt + b2);                  // cos(atan2(|cross|, dot))

  float c2 = c*c, c3 = c2*c, c4 = c2*c2, c5 = c4*c, c6 = c4*c2;
  // A (16x4 f32 per WMMA): VGPR0 = K{0|2}, VGPR1 = K{1|3}; hi lanes carry K+2
  v2f A0, A1, B0, B1;
  A0.x = hi ? c2 : 1.0f;  A0.y = hi ? c3 : c;
  A1.x = hi ? c6 : c4;    A1.y = hi ? 0.0f : c5;       // K=7 row of B is zero
  B0.x = BM[hi ? 2 : 0][tl];  B0.y = BM[hi ? 3 : 1][tl];
  B1.x = BM[hi ? 6 : 4][tl];  B1.y = BM[hi ? 7 : 5][tl];

  v8f acc = {};
  acc = __builtin_amdgcn_wmma_f32_16x16x4_f32(false, A0, false, B0, (short)0, acc, false, false);
  acc = __builtin_amdgcn_wmma_f32_16x16x4_f32(false, A1, false, B1, (short)0, acc, false, false);

  // D layout: VGPR r = row M=r (lanes 0-15) / M=r+8 (lanes 16-31); column N = tl
  if (tl < 7) {
    #pragma unroll
    for (int r = 0; r < 8; ++r) cbf_s[wave][hi * 8 + r][tl] = acc[r];
  }
  if (!hi) kj_s[wave][tl] = e;
  __syncthreads();

  // 16 triplets * 42 = 672 outputs per wave-tile: flat loop -> contiguous stores
  size_t obase = (size_t)t0 * 42;
  #pragma unroll
  for (int w = 0; w < 21; ++w) {
    int flat = w * 32 + lane;            // 0..671
    int t2   = flat / 42;                // local triplet
    int cc   = flat - t2 * 42;           // 0..41 = l*6+n
    int l    = cc / 6;
    int e2   = kj_s[wave][t2];
    angle_out[obase + flat] = rbf_ws[(size_t)e2 * 42 + cc] * cbf_s[wave][t2][l];
  }
}

extern "C" void kernel_launch(void* const* d_in, const int* in_sizes, int n_in,
                              void* d_out, int out_size, void* d_ws, size_t ws_size,
                              hipStream_t stream) {
  const float* coords = (const float*)d_in[0];
  const float* freq   = (const float*)d_in[1];
  const int* nis = (const int*)d_in[2];
  const int* njs = (const int*)d_in[3];
  const int* ti  = (const int*)d_in[4];
  const int* tj  = (const int*)d_in[5];
  const int* tk  = (const int*)d_in[6];
  const int* kjp = (const int*)d_in[7];

  float* out       = (float*)d_out;
  float* dist_out  = out;                               // [E,6]
  float* angle_out = out + (size_t)N_EDGES * 6;         // [T,42]

  float* normw  = (float*)d_ws;                         // 42 floats
  float* rbf_ws = (float*)((char*)d_ws + 256);          // [E][42] floats (~336 MB)

  norm_kernel<<<1, 64, 0, stream>>>(normw);
  edge_kernel<<<(N_EDGES + 255) / 256, 256, 0, stream>>>(coords, freq, nis, njs, normw,
                                                         dist_out, rbf_ws);
  trip_kernel<<<N_TRIP / 128, 256, 0, stream>>>(coords, ti, tj, tk, kjp, rbf_ws, angle_out);
}